// SampledSoftmaxLoss_32667521254266
// MI455X (gfx1250) — compile-verified
//
#include <hip/hip_runtime.h>
#include <hip/hip_bf16.h>
#include <math.h>

typedef __attribute__((ext_vector_type(2))) float v2f;
typedef __attribute__((ext_vector_type(8))) float v8f;

#define DIM 64
#define KNEG 100
#define NSLOTS 101            // slot 0 = positive, 1..100 = negatives
#define ROWS 16               // n-rows per block (one WMMA tile of columns)
#define NWAVES 8              // 256 threads / wave32
#define TEMP_INV 20.0f        // 1 / 0.05
#define EPS_NRM 1e-12f

// ---------------------------------------------------------------------------
// Kernel 1: per-item inverse L2 norms (table is 25.6MB -> one L2-resident pass)
// 16 lanes per row, each lane loads a float4 (16 lanes * 16B = 256B row).
// ---------------------------------------------------------------------------
__global__ __launch_bounds__(256)
void item_inv_norms_kernel(const float* __restrict__ items,
                           float* __restrict__ invn, int V) {
    int tid    = threadIdx.x;
    int row    = blockIdx.x * 16 + (tid >> 4);
    int lane16 = tid & 15;
    if (row >= V) return;
    const float4 v = *(const float4*)(items + (size_t)row * DIM + lane16 * 4);
    float s = v.x * v.x + v.y * v.y + v.z * v.z + v.w * v.w;
    // xor-reduce within the 16-lane group (wave32)
    s += __shfl_xor(s, 1, 32);
    s += __shfl_xor(s, 2, 32);
    s += __shfl_xor(s, 4, 32);
    s += __shfl_xor(s, 8, 32);
    if (lane16 == 0) invn[row] = 1.0f / fmaxf(sqrtf(s), EPS_NRM);
}

// ---------------------------------------------------------------------------
// Kernel 2: zero the two global accumulators (num, den)
// ---------------------------------------------------------------------------
__global__ void zero_accum_kernel(float* __restrict__ accum) {
    accum[0] = 0.0f;
    accum[1] = 0.0f;
}

// ---------------------------------------------------------------------------
// Kernel 3: main sampled-softmax loss.
// Block = 256 threads = 8 waves, handles ROWS=16 consecutive n-rows.
//  - Stage A: load+normalize the 16x64 vo tile in LDS.
//  - Stage B: each wave owns logit slots {wv, wv+8, ...}. Per slot it gathers
//    16 item rows as WMMA A (f32 16x4 layout: lane m / m+16 each load a
//    float2 per K-chunk), B = vo columns from LDS, and runs a 16-step
//    v_wmma_f32_16x16x4_f32 chain over K=64. Diagonal of the 16x16 D tile
//    = the 16 logits; scale by item inv-norm and 1/TEMP.
//  - Stage C: per-row logsumexp over 101 logits, weighted block reduction,
//    two global f32 atomics.
// ---------------------------------------------------------------------------
__global__ __launch_bounds__(256)
void sampled_softmax_main_kernel(const float* __restrict__ out_emb,
                                 const int*   __restrict__ target,
                                 const float* __restrict__ items,
                                 const float* __restrict__ weights,
                                 const int*   __restrict__ negs,
                                 const int*   __restrict__ alts,
                                 const float* __restrict__ item_invn,
                                 float*       __restrict__ accum) {
    __shared__ float lds_vo[ROWS][DIM];              // 4 KB, normalized vo tile
    __shared__ float lds_invvo[ROWS];
    __shared__ float lds_tile[NWAVES][16][16];       // 8 KB, per-wave D tiles
    __shared__ float lds_logits[ROWS][NSLOTS + 3];   // padded stride 104
    __shared__ float lds_red[2][ROWS];

    const int tid = threadIdx.x;
    const int n0  = blockIdx.x * ROWS;

    // ---- Stage A: vo tile load (coalesced float4) + in-LDS normalize ----
    {
        float4 v = *(const float4*)(out_emb + (size_t)n0 * DIM + tid * 4);
        *(float4*)(&lds_vo[0][0] + tid * 4) = v;
    }
    __syncthreads();
    if (tid < ROWS) {
        float s = 0.0f;
        #pragma unroll
        for (int c = 0; c < DIM; ++c) { float x = lds_vo[tid][c]; s += x * x; }
        lds_invvo[tid] = 1.0f / fmaxf(sqrtf(s), EPS_NRM);
    }
    __syncthreads();
    {
        int r = (tid * 4) / DIM;
        float inv = lds_invvo[r];
        float4* p = (float4*)(&lds_vo[0][0] + tid * 4);
        float4 v = *p;
        v.x *= inv; v.y *= inv; v.z *= inv; v.w *= inv;
        *p = v;
    }
    __syncthreads();

    // ---- Stage B: WMMA logit slots ----
    const int wv   = tid >> 5;     // wave id 0..7 (uniform per wave)
    const int lane = tid & 31;
    const int m    = lane & 15;    // tile row / column index
    const int hi   = lane >> 4;    // 0: lanes 0-15, 1: lanes 16-31
    const int n    = n0 + m;
    const int tgt  = target[n];

    for (int k = wv; k < NSLOTS; k += NWAVES) {   // wave-uniform loop bound
        int idx;
        if (k == 0) {
            idx = tgt;                             // positive slot
        } else {
            int j = k - 1;
            idx = negs[(size_t)n * KNEG + j];
            if (idx == tgt) idx = alts[(size_t)n * KNEG + j];  // resample
        }
        const float* erow = items + (size_t)idx * DIM;

        v8f acc = {};
        #pragma unroll
        for (int kk = 0; kk < 16; ++kk) {
            const int c0 = kk * 4 + hi * 2;        // K offset for this lane half
            // A: 16x4 f32 tile, row M=m gathered from item table (8B load)
            const float2 av = *(const float2*)(erow + c0);
            // B: 4x16 f32 tile, column N=m from normalized vo in LDS
            const float2 bv = *(const float2*)(&lds_vo[m][c0]);
            v2f a; a.x = av.x; a.y = av.y;
            v2f b; b.x = bv.x; b.y = bv.y;
            acc = __builtin_amdgcn_wmma_f32_16x16x4_f32(
                false, a, false, b, (short)0, acc, false, false);
        }

        // dump D tile: lane covers (M = v + hi*8, N = m) for VGPR v
        #pragma unroll
        for (int v = 0; v < 8; ++v)
            lds_tile[wv][v + hi * 8][m] = acc[v];
        // cross-lane LDS dependency inside one wave: wait on DScnt explicitly
        asm volatile("s_wait_dscnt 0" ::: "memory");

        if (hi == 0) {
            float logit = lds_tile[wv][m][m] * item_invn[idx] * TEMP_INV;
            lds_logits[m][k] = logit;
        }
    }
    __syncthreads();

    // ---- Stage C: per-row logsumexp + weighted reduction ----
    if (tid < ROWS) {
        float mx = -3.4e38f;
        for (int k = 0; k < NSLOTS; ++k) mx = fmaxf(mx, lds_logits[tid][k]);
        float s = 0.0f;
        for (int k = 0; k < NSLOTS; ++k) s += __expf(lds_logits[tid][k] - mx);
        float lse  = mx + __logf(s);
        float loss = lse - lds_logits[tid][0];
        float w    = weights[n0 + tid];
        float wm   = (w > 0.0f) ? w : 0.0f;
        lds_red[0][tid] = loss * wm;
        lds_red[1][tid] = wm;
    }
    __syncthreads();
    if (tid == 0) {
        float sn = 0.0f, sd = 0.0f;
        #pragma unroll
        for (int i = 0; i < ROWS; ++i) { sn += lds_red[0][i]; sd += lds_red[1][i]; }
        atomicAdd(&accum[0], sn);
        atomicAdd(&accum[1], sd);
    }
}

// ---------------------------------------------------------------------------
// Kernel 4: final scalar divide
// ---------------------------------------------------------------------------
__global__ void finalize_kernel(const float* __restrict__ accum,
                                float* __restrict__ out) {
    out[0] = accum[0] / accum[1];
}

// ---------------------------------------------------------------------------
extern "C" void kernel_launch(void* const* d_in, const int* in_sizes, int n_in,
                              void* d_out, int out_size, void* d_ws, size_t ws_size,
                              hipStream_t stream) {
    const float* out_emb = (const float*)d_in[0];   // [N, 64]
    const int*   target  = (const int*)  d_in[1];   // [N]
    const float* items   = (const float*)d_in[2];   // [V, 64]
    const float* weights = (const float*)d_in[3];   // [N]
    const int*   negs    = (const int*)  d_in[4];   // [N, 100]
    const int*   alts    = (const int*)  d_in[5];   // [N, 100]

    const int N = in_sizes[1];          // 51200
    const int V = in_sizes[2] / DIM;    // 100001

    float* accum = (float*)d_ws;            // 2 floats: num, den
    float* invn  = (float*)d_ws + 16;       // V floats (64B-offset)

    zero_accum_kernel<<<1, 1, 0, stream>>>(accum);
    item_inv_norms_kernel<<<(V + 15) / 16, 256, 0, stream>>>(items, invn, V);
    sampled_softmax_main_kernel<<<N / ROWS, 256, 0, stream>>>(
        out_emb, target, items, weights, negs, alts, invn, accum);
    finalize_kernel<<<1, 1, 0, stream>>>(accum, (float*)d_out);
}